// QuantumLayer_50964081934559
// MI455X (gfx1250) — compile-verified
//
#include <hip/hip_runtime.h>

typedef float v2f __attribute__((ext_vector_type(2)));
typedef float v8f __attribute__((ext_vector_type(8)));

static constexpr int    Dq = 1024;   // D
static constexpr int    Rq = 100;    // measurement rounds
static constexpr int    Bq = 16;     // batch
static constexpr size_t DD = (size_t)Dq * (size_t)Dq;

// ---------------------------------------------------------------------------
// Stage 0: Ccos = cos(phase), vectorized x4 (1M elems, pure bandwidth)
// ---------------------------------------------------------------------------
__global__ __launch_bounds__(256) void cos_kernel(const float* __restrict__ ph,
                                                  float* __restrict__ cc) {
    int t = blockIdx.x * blockDim.x + threadIdx.x;        // 0 .. DD/4-1
    float4 p = ((const float4*)ph)[t];
    float4 c;
    c.x = cosf(p.x); c.y = cosf(p.y); c.z = cosf(p.z); c.w = cosf(p.w);
    ((float4*)cc)[t] = c;
}

// ---------------------------------------------------------------------------
// Stage 1a: Gsum = mean_r gate_W[r]  (400 MB read — the dominant cost)
// ---------------------------------------------------------------------------
__global__ __launch_bounds__(256) void reduce_gw(const float* __restrict__ gW,
                                                 float* __restrict__ gs) {
    int t = blockIdx.x * blockDim.x + threadIdx.x;        // 0 .. DD/4-1
    const float4* src = (const float4*)gW;
    const size_t  stride = DD / 4;
    float4 s = {0.f, 0.f, 0.f, 0.f};
    for (int r = 0; r < Rq; ++r) {
        float4 v = src[(size_t)r * stride + t];
        s.x += v.x; s.y += v.y; s.z += v.z; s.w += v.w;
    }
    const float inv = 1.0f / (float)Rq;
    float4 o; o.x = s.x * inv; o.y = s.y * inv; o.z = s.z * inv; o.w = s.w * inv;
    ((float4*)gs)[t] = o;
}

// ---------------------------------------------------------------------------
// Stage 1b: gb_mean = mean_r gate_b[r]
// ---------------------------------------------------------------------------
__global__ __launch_bounds__(256) void reduce_gb(const float* __restrict__ gb,
                                                 float* __restrict__ gm) {
    int k = blockIdx.x * blockDim.x + threadIdx.x;        // 0 .. D-1
    float s = 0.f;
    for (int r = 0; r < Rq; ++r) s += gb[r * Dq + k];
    gm[k] = s * (1.0f / (float)Rq);
}

// ---------------------------------------------------------------------------
// fp32 WMMA GEMM:  Cout(z) = A @ B'  (+ bias), one 16x16 tile per wave32.
//   TRANS_B : B supplied row-major N x K  (GEMM-NT, contiguous float2 loads)
//   SCALE   : B'[j][n] = scale[z][j] * B[j][n]   (per-K row scaling)
//   BIAS    : add bias[col] to every output element
// Block = 256 threads = 8 waves tiled 2(M) x 4(N) -> 32 x 64 block tile.
// A 16x4 frag : lane<16 -> K pair {k,k+1}, lane>=16 -> {k+2,k+3}, M = lane&15
// B 4x16 frag : same K split, N = lane&15
// C/D         : VGPR v -> row v + 8*(lane>=16), col lane&15
// ---------------------------------------------------------------------------
template <bool TRANS_B, bool SCALE, bool BIAS>
__global__ __launch_bounds__(256) void wmma_gemm(const float* __restrict__ A,
                                                 const float* __restrict__ Bm,
                                                 const float* __restrict__ scaleBase,
                                                 const float* __restrict__ bias,
                                                 float* __restrict__ Cout) {
    const int lane = threadIdx.x & 31;
    const int wave = threadIdx.x >> 5;
    const int wm   = wave >> 2;                 // 0..1
    const int wn   = wave & 3;                  // 0..3
    const int row0 = blockIdx.y * 32 + wm * 16;
    const int col0 = blockIdx.x * 64 + wn * 16;
    const int half = lane >> 4;                 // 0 or 1
    const int mn   = lane & 15;                 // M for A-frag, N for B/C frags

    const float* scale = SCALE ? (scaleBase + (size_t)blockIdx.z * Dq) : nullptr;
    float* outp = Cout + (size_t)blockIdx.z * DD;

    const float* arow = A + (size_t)(row0 + mn) * Dq;   // A row for this lane
    const float* bcol;                                   // B addressing base
    if (TRANS_B) bcol = Bm + (size_t)(col0 + mn) * Dq;   // row of B^T source
    else         bcol = Bm + (size_t)(col0 + mn);        // column of B

    v8f acc = {0.f, 0.f, 0.f, 0.f, 0.f, 0.f, 0.f, 0.f};

#pragma unroll 4
    for (int k = 0; k < Dq; k += 4) {
        const int ka = k + 2 * half;
        v2f a = *(const v2f*)(arow + ka);
        v2f b;
        if (TRANS_B) {
            b = *(const v2f*)(bcol + ka);                 // contiguous (NT)
        } else {
            b.x = bcol[(size_t)ka * Dq];
            b.y = bcol[(size_t)(ka + 1) * Dq];
            if (SCALE) { b.x *= scale[ka]; b.y *= scale[ka + 1]; }
        }
        // D = A(16x4) * B(4x16) + C   -> v_wmma_f32_16x16x4_f32
        acc = __builtin_amdgcn_wmma_f32_16x16x4_f32(
            /*neg_a=*/false, a, /*neg_b=*/false, b,
            /*c_mod=*/(short)0, acc, /*reuse_a=*/false, /*reuse_b=*/false);
    }

    const float bv = BIAS ? bias[col0 + mn] : 0.0f;
#pragma unroll
    for (int v = 0; v < 8; ++v) {
        outp[(size_t)(row0 + v + 8 * half) * Dq + (col0 + mn)] = acc[v] + bv;
    }
}

// ---------------------------------------------------------------------------
// Launch:  d_in = { x(16,1,1024), QW(DxD), phase(DxD), EM(DxD),
//                   gate_W(R,D,D), gate_b(R,D) }   all fp32
// d_out = out(16, 1024, 1024) fp32
// d_ws  : Gsum | T1 | M | Ccos | gb_mean  (4*DD + D floats ~= 16.8 MB)
// ---------------------------------------------------------------------------
extern "C" void kernel_launch(void* const* d_in, const int* in_sizes, int n_in,
                              void* d_out, int out_size, void* d_ws, size_t ws_size,
                              hipStream_t stream) {
    const float* x  = (const float*)d_in[0];
    const float* qw = (const float*)d_in[1];
    const float* ph = (const float*)d_in[2];
    const float* em = (const float*)d_in[3];
    const float* gW = (const float*)d_in[4];
    const float* gb = (const float*)d_in[5];
    float* out = (float*)d_out;

    float* f    = (float*)d_ws;
    float* Gsum = f;            // DD
    float* T1   = f + DD;       // DD
    float* Mm   = f + 2 * DD;   // DD
    float* Cc   = f + 3 * DD;   // DD
    float* gbm  = f + 4 * DD;   // D

    const dim3 blk(256);
    const dim3 grid_bw(DD / 4 / 256);          // 1024 blocks, x4-vectorized
    const dim3 grid_gemm(Dq / 64, Dq / 32, 1); // 16 x 32 (32x64 block tiles)
    const dim3 grid_out(Dq / 64, Dq / 32, Bq); // + batch in z

    // Stage 0/1: bandwidth kernels (gate_W reduction dominates: ~400 MB read)
    cos_kernel<<<grid_bw, blk, 0, stream>>>(ph, Cc);
    reduce_gw <<<grid_bw, blk, 0, stream>>>(gW, Gsum);
    reduce_gb <<<dim3(Dq / 256), blk, 0, stream>>>(gb, gbm);

    // Stage 2: T1 = QW @ EM                       (NN)
    wmma_gemm<false, false, false><<<grid_gemm, blk, 0, stream>>>(qw, em, nullptr, nullptr, T1);
    // Stage 3: M = T1 @ Gsum^T                    (NT)
    wmma_gemm<true,  false, false><<<grid_gemm, blk, 0, stream>>>(T1, Gsum, nullptr, nullptr, Mm);
    // Stage 4: out[b] = Ccos @ (diag(x_b) @ M) + gb_mean   (NN + row-scale + bias)
    wmma_gemm<false, true,  true ><<<grid_out,  blk, 0, stream>>>(Cc, Mm, x, gbm, out);
}